// PairWiseLoss_10359461118671
// MI455X (gfx1250) — compile-verified
//
#include <hip/hip_runtime.h>
#include <math.h>

// Pairwise ranking loss: loss = -(1/C) * sum_{i<j} log_sigmoid(p_i - p_j), n=8192.
// 16x16 tiles of the strict upper triangle. Tile difference matrix
// D[i][j] = p_row[i] - p_col[j] produced by one V_WMMA_F32_16X16X4_F32
// (rank-2 outer product: A=[p_row, 1, 0, 0], B=[1; -p_col; 0; 0]).
// log_sigmoid:  logsig(x) = min(x,0) - ln2*log2(1 + 2^(-|x|*log2e)).
// TRANS batching: sum_v log2(1+e_v) = log2(prod_v (1+e_v)); factors in [1,2],
// product over 8 elements <= 256, so one v_log_f32 per 8 v_exp_f32.
// Off-diagonal interior tiles take an unmasked fast path; diagonal/tail tiles
// map masked elements to x=+inf -> factor 1 / min 0 -> exact zero contribution.
// Deterministic multi-stage reductions (no atomics).

typedef float v2f __attribute__((ext_vector_type(2)));
typedef float v8f __attribute__((ext_vector_type(8)));

#define MAX_N   8192
#define TILE    16
#define BLOCK   256   // 8 wave32 per block

#define LOG2E_F 1.44269504088896340736f
#define LN2_F   0.69314718055994530942f

__global__ void pairwise_tiles_kernel(const float* __restrict__ pred,
                                      float* __restrict__ partials,
                                      int n) {
    __shared__ float p[MAX_N];
    __shared__ float wsum[BLOCK / 32];

    // Stage entire prediction vector in LDS (32 KB of the 320 KB WGP LDS).
    for (int i = threadIdx.x; i < MAX_N; i += blockDim.x)
        p[i] = (i < n) ? pred[i] : 0.0f;
    __syncthreads();

    const int lane          = threadIdx.x & 31;
    const int wave_in_blk   = threadIdx.x >> 5;
    const int waves_per_blk = blockDim.x >> 5;
    const int gwave  = blockIdx.x * waves_per_blk + wave_in_blk;
    const int nwaves = gridDim.x * waves_per_blk;

    const bool lo16  = (lane < 16);
    const int  nloc  = lane & 15;            // A row index / B-C-D column index
    const int  mbase = (lane >> 4) << 3;     // C/D rows: 0..7 (lanes 0-15), 8..15 (lanes 16-31)

    const int nt     = (n + TILE - 1) / TILE;   // tiles per dim
    const int ntiles = nt * (nt + 1) / 2;       // upper triangle incl. diagonal

    float acc_lin = 0.0f;   // sum of min(x,0)
    float acc_l2  = 0.0f;   // sum of log2(1 + 2^(-|x|*log2e))

    for (int t = gwave; t < ntiles; t += nwaves) {
        // Decode (ti, tj), ti <= tj, row-major over the upper triangle:
        // base(ti) = ti*nt - ti*(ti-1)/2
        double bq = 2.0 * nt + 1.0;
        int ti = (int)floor((bq - sqrt(bq * bq - 8.0 * (double)t)) * 0.5);
        while (ti > 0 && ti * nt - ti * (ti - 1) / 2 > t) --ti;
        while ((ti + 1) * nt - (ti + 1) * ti / 2 <= t) ++ti;
        const int base = ti * nt - ti * (ti - 1) / 2;
        const int tj = ti + (t - base);
        const int i0 = ti * TILE;
        const int j0 = tj * TILE;

        // A (16x4): VGPR0 = K=0 (lanes 0-15) / K=2 (lanes 16-31), VGPR1 = K=1 / K=3.
        // A[:,0] = p_row, A[:,1] = 1, A[:,2..3] = 0.
        v2f a;
        a.x = lo16 ? p[i0 + nloc] : 0.0f;
        a.y = lo16 ? 1.0f : 0.0f;
        // B (4x16): VGPR0 = K=0 (lanes 0-15) / K=2 (lanes 16-31), VGPR1 = K=1 / K=3.
        // B[0,:] = 1, B[1,:] = -p_col, B[2..3,:] = 0.
        v2f b;
        b.x = lo16 ? 1.0f : 0.0f;
        b.y = lo16 ? -p[j0 + nloc] : 0.0f;

        v8f c = {0.f, 0.f, 0.f, 0.f, 0.f, 0.f, 0.f, 0.f};
        // D[i][j] = p_row[i] - p_col[j]
        c = __builtin_amdgcn_wmma_f32_16x16x4_f32(
                /*neg_a=*/false, a, /*neg_b=*/false, b,
                /*c_mod=*/(short)0, c, /*reuse_a=*/false, /*reuse_b=*/false);

        // Wave-uniform tile classification: off-diagonal fully-interior tiles
        // need no element mask (~99.6% of tiles for n=8192).
        if (ti != tj && j0 + TILE <= n) {
            float prod = 1.0f;
            #pragma unroll
            for (int v = 0; v < 8; ++v) {
                const float x  = c[v];
                const float e2 = __builtin_amdgcn_exp2f(fabsf(x) * -LOG2E_F);
                prod    *= (1.0f + e2);        // factor in [1,2]
                acc_lin += fminf(x, 0.0f);
            }
            acc_l2 += __builtin_amdgcn_logf(prod);   // one v_log_f32 per 8 elems
        } else {
            const int jg = j0 + nloc;        // global column index for this lane
            float prod = 1.0f;
            #pragma unroll
            for (int v = 0; v < 8; ++v) {
                const int ig = i0 + mbase + v;   // global row index for c[v]
                const bool keep = (jg > ig) && (jg < n);
                // masked-out -> +inf: exp2(-inf)=0 -> factor 1; min(inf,0)=0
                const float x  = keep ? c[v] : __builtin_inff();
                const float e2 = __builtin_amdgcn_exp2f(fabsf(x) * -LOG2E_F);
                prod    *= (1.0f + e2);
                acc_lin += fminf(x, 0.0f);
            }
            acc_l2 += __builtin_amdgcn_logf(prod);
        }
    }

    float acc = acc_lin - LN2_F * acc_l2;

    // wave32 xor-tree reduction (deterministic)
    #pragma unroll
    for (int off = 16; off > 0; off >>= 1)
        acc += __shfl_xor(acc, off, 32);

    if (lane == 0) wsum[wave_in_blk] = acc;
    __syncthreads();
    if (threadIdx.x == 0) {
        float s = 0.0f;
        for (int w = 0; w < waves_per_blk; ++w) s += wsum[w];
        partials[blockIdx.x] = s;
    }
}

__global__ void final_reduce_kernel(const float* __restrict__ partials,
                                    int nparts, int n,
                                    float* __restrict__ out) {
    __shared__ double s[BLOCK];
    double acc = 0.0;
    for (int i = threadIdx.x; i < nparts; i += blockDim.x)
        acc += (double)partials[i];
    s[threadIdx.x] = acc;
    __syncthreads();
    if (threadIdx.x == 0) {
        double tot = 0.0;
        for (int i = 0; i < BLOCK; ++i) tot += s[i];   // fixed order: deterministic
        const double cnt = (double)n * (double)(n - 1) * 0.5;
        out[0] = (float)(-tot / cnt);
    }
}

extern "C" void kernel_launch(void* const* d_in, const int* in_sizes, int n_in,
                              void* d_out, int out_size, void* d_ws, size_t ws_size,
                              hipStream_t stream) {
    (void)n_in; (void)out_size;
    const float* pred = (const float*)d_in[0];
    float* out = (float*)d_out;
    const int n = in_sizes[0];

    // Per-block partials live in workspace; clamp grid to workspace capacity.
    int nblocks = 1024;
    size_t cap = ws_size / sizeof(float);
    if (cap < (size_t)nblocks) nblocks = (int)cap;

    float* partials = (float*)d_ws;
    if (nblocks < 1) {             // degenerate workspace: stage partial in d_out
        nblocks = 1;
        partials = out;
    }

    pairwise_tiles_kernel<<<nblocks, BLOCK, 0, stream>>>(pred, partials, n);
    final_reduce_kernel<<<1, BLOCK, 0, stream>>>(partials, nblocks, n, out);
}